// DiscreteModel_58033598104174
// MI455X (gfx1250) — compile-verified
//
#include <hip/hip_runtime.h>
#include <hip/hip_bf16.h>
#include <math.h>

// Sizes (compile-time constants from the reference)
#define NN   4096
#define MD   64
#define MSG  64
#define RAW  4224      // MD + NN + MSG
#define HID  2112      // RAW/2
#define WL   8

typedef __attribute__((ext_vector_type(2))) float v2f;
typedef __attribute__((ext_vector_type(8))) float v8f;

__device__ __forceinline__ v2f ldv2(const float* p) { return *(const v2f*)p; }

// One K-step (K += 4) of a 4x4-tile (64x64) WMMA block.
// A row-major [.., ldA], B row-major [.., ldB]; computes C += A * B^T tilewise.
__device__ __forceinline__ void wmma_step4x4(
    v8f acc[4][4],
    const float* __restrict__ A, int r0, int ldA, int kA,
    const float* __restrict__ B, int c0, int ldB, int kB, int half) {
  v2f a[4], b[4];
#pragma unroll
  for (int t = 0; t < 4; ++t) {
    a[t] = ldv2(A + (size_t)(r0 + 16 * t) * (size_t)ldA + kA + 2 * half);
    b[t] = ldv2(B + (size_t)(c0 + 16 * t) * (size_t)ldB + kB + 2 * half);
  }
#pragma unroll
  for (int mi = 0; mi < 4; ++mi)
#pragma unroll
    for (int ni = 0; ni < 4; ++ni)
      acc[mi][ni] = __builtin_amdgcn_wmma_f32_16x16x4_f32(
          false, a[mi], false, b[ni], (short)0, acc[mi][ni], false, false);
}

// ---------------------------------------------------------------------------
// K1: random-walk pooled embedding:  rw = mean(memory[walks],axis=1) @ W_rw^T + b_rw
// grid: NN blocks of 64 threads
__global__ __launch_bounds__(64) void k_rw(
    const float* __restrict__ memory, const int* __restrict__ walks,
    const float* __restrict__ W_rw, const float* __restrict__ b_rw,
    float* __restrict__ rw) {
  __shared__ float pooled[MD];
  const int n = blockIdx.x;
  const int d = threadIdx.x;
  float s = 0.f;
#pragma unroll
  for (int w = 0; w < WL; ++w) {
    const int idx = walks[n * WL + w];
    s += memory[(size_t)idx * MD + d];
  }
  pooled[d] = s * (1.0f / WL);
  __syncthreads();
  float acc = b_rw[d];
#pragma unroll 8
  for (int k = 0; k < MD; ++k) acc += pooled[k] * W_rw[d * MD + k];
  rw[(size_t)n * MSG + d] = acc;
}

// ---------------------------------------------------------------------------
// K2: hid = relu(concat(memory, od, rw) @ W1^T + b1)   [NN x HID]
// grid: (HID/64, NN/64), block: 32 (one wave -> 64x64 C tile)
__global__ __launch_bounds__(32) void k_gemm1(
    const float* __restrict__ memory, const float* __restrict__ od,
    const float* __restrict__ rw, const float* __restrict__ W1,
    const float* __restrict__ bias1, float* __restrict__ hid) {
  const int lane = threadIdx.x & 31;
  const int l15 = lane & 15;
  const int half = lane >> 4;
  const int nBase = blockIdx.x * 64;   // HID cols
  const int mBase = blockIdx.y * 64;   // node rows
  const int r0 = mBase + l15;
  const int c0 = nBase + l15;

  v8f acc[4][4] = {};

  // segment 1: k in [0,64) comes from memory (ld=MD)
  for (int k0 = 0; k0 < MD; k0 += 4)
    wmma_step4x4(acc, memory, r0, MD, k0, W1, c0, RAW, k0, half);
  // segment 2: k in [64, 64+NN) comes from od_mat (ld=NN)
  for (int k0 = MD; k0 < MD + NN; k0 += 4)
    wmma_step4x4(acc, od, r0, NN, k0 - MD, W1, c0, RAW, k0, half);
  // segment 3: k in [64+NN, RAW) comes from rw (ld=MSG)
  for (int k0 = MD + NN; k0 < RAW; k0 += 4)
    wmma_step4x4(acc, rw, r0, MSG, k0 - (MD + NN), W1, c0, RAW, k0, half);

#pragma unroll
  for (int ni = 0; ni < 4; ++ni) {
    const int col = nBase + ni * 16 + l15;
    const float bv = bias1[col];
#pragma unroll
    for (int mi = 0; mi < 4; ++mi) {
#pragma unroll
      for (int i = 0; i < 8; ++i) {
        const int row = mBase + mi * 16 + i + 8 * half;
        const float v = acc[mi][ni][i] + bv;
        hid[(size_t)row * HID + col] = v > 0.f ? v : 0.f;
      }
    }
  }
}

// ---------------------------------------------------------------------------
// K3: msg = hid @ W2^T + b2   [NN x MSG], one wave does a 16x64 C strip
// grid: NN/16 blocks of 32
__global__ __launch_bounds__(32) void k_gemm2(
    const float* __restrict__ hid, const float* __restrict__ W2,
    const float* __restrict__ bias2, float* __restrict__ msg) {
  const int lane = threadIdx.x & 31;
  const int l15 = lane & 15;
  const int half = lane >> 4;
  const int mBase = blockIdx.x * 16;
  const int r0 = mBase + l15;

  v8f acc[4] = {};
  for (int k0 = 0; k0 < HID; k0 += 4) {
    const v2f a = ldv2(hid + (size_t)r0 * HID + k0 + 2 * half);
#pragma unroll
    for (int ni = 0; ni < 4; ++ni) {
      const v2f b = ldv2(W2 + (size_t)(ni * 16 + l15) * HID + k0 + 2 * half);
      acc[ni] = __builtin_amdgcn_wmma_f32_16x16x4_f32(
          false, a, false, b, (short)0, acc[ni], false, false);
    }
  }
#pragma unroll
  for (int ni = 0; ni < 4; ++ni) {
    const int col = ni * 16 + l15;
    const float bv = bias2[col];
#pragma unroll
    for (int i = 0; i < 8; ++i) {
      const int row = mBase + i + 8 * half;
      msg[(size_t)row * MSG + col] = acc[ni][i] + bv;
    }
  }
}

// ---------------------------------------------------------------------------
// K4: fused GRUCell update -> upd [NN x MD]
// grid: NN blocks of 64 threads (thread j = output dim)
__global__ __launch_bounds__(64) void k_gru(
    const float* __restrict__ msg, const float* __restrict__ memory,
    const float* __restrict__ Wi, const float* __restrict__ bi,
    const float* __restrict__ Wh, const float* __restrict__ bh,
    float* __restrict__ upd) {
  __shared__ float sm[MSG];
  __shared__ float sh[MD];
  const int n = blockIdx.x;
  const int j = threadIdx.x;
  sm[j] = msg[(size_t)n * MSG + j];
  sh[j] = memory[(size_t)n * MD + j];
  __syncthreads();
  float ir = bi[j], iz = bi[MD + j], in_ = bi[2 * MD + j];
  float hr = bh[j], hz = bh[MD + j], hn = bh[2 * MD + j];
#pragma unroll 4
  for (int m = 0; m < MSG; ++m) {
    const float mv = sm[m];
    const float hv = sh[m];
    ir += mv * Wi[(size_t)(j) * MSG + m];
    iz += mv * Wi[(size_t)(MD + j) * MSG + m];
    in_ += mv * Wi[(size_t)(2 * MD + j) * MSG + m];
    hr += hv * Wh[(size_t)(j) * MD + m];
    hz += hv * Wh[(size_t)(MD + j) * MD + m];
    hn += hv * Wh[(size_t)(2 * MD + j) * MD + m];
  }
  const float r = 1.f / (1.f + __expf(-(ir + hr)));
  const float z = 1.f / (1.f + __expf(-(iz + hz)));
  const float nn = tanhf(in_ + r * hn);
  upd[(size_t)n * MD + j] = (1.f - z) * nn + z * sh[j];
}

// ---------------------------------------------------------------------------
// K5: p = relu(upd @ Wp1^T + bp1)  [NN x MD]
// grid: NN blocks of 64 threads
__global__ __launch_bounds__(64) void k_pred1(
    const float* __restrict__ upd, const float* __restrict__ Wp1,
    const float* __restrict__ bp1, float* __restrict__ p) {
  __shared__ float su[MD];
  const int n = blockIdx.x;
  const int j = threadIdx.x;
  su[j] = upd[(size_t)n * MD + j];
  __syncthreads();
  float acc = bp1[j];
#pragma unroll 8
  for (int d = 0; d < MD; ++d) acc += su[d] * Wp1[j * MD + d];
  p[(size_t)n * MD + j] = acc > 0.f ? acc : 0.f;
}

// ---------------------------------------------------------------------------
// K6: od_out = p @ Wp2^T + bp2  [NN x NN], K = MD
// grid: (NN/256, NN/64), block 128 (4 waves, each a 64x64 C tile)
__global__ __launch_bounds__(128) void k_pred2(
    const float* __restrict__ p, const float* __restrict__ Wp2,
    const float* __restrict__ bp2, float* __restrict__ out) {
  const int lane = threadIdx.x & 31;
  const int wave = threadIdx.x >> 5;
  const int l15 = lane & 15;
  const int half = lane >> 4;
  const int mBase = blockIdx.y * 64;
  const int nBase = blockIdx.x * 256 + wave * 64;
  const int r0 = mBase + l15;
  const int c0 = nBase + l15;

  v8f acc[4][4] = {};
#pragma unroll
  for (int k0 = 0; k0 < MD; k0 += 4)
    wmma_step4x4(acc, p, r0, MD, k0, Wp2, c0, MD, k0, half);

#pragma unroll
  for (int ni = 0; ni < 4; ++ni) {
    const int col = nBase + ni * 16 + l15;
    const float bv = bp2[col];
#pragma unroll
    for (int mi = 0; mi < 4; ++mi) {
#pragma unroll
      for (int i = 0; i < 8; ++i) {
        const int row = mBase + mi * 16 + i + 8 * half;
        out[(size_t)row * NN + col] = acc[mi][ni][i] + bv;
      }
    }
  }
}

// ---------------------------------------------------------------------------
extern "C" void kernel_launch(void* const* d_in, const int* in_sizes, int n_in,
                              void* d_out, int out_size, void* d_ws, size_t ws_size,
                              hipStream_t stream) {
  const float* memory = (const float*)d_in[0];
  const float* od_mat = (const float*)d_in[1];
  const int*   walks  = (const int*)d_in[2];
  const float* W_rw   = (const float*)d_in[3];
  const float* b_rw   = (const float*)d_in[4];
  const float* W1     = (const float*)d_in[5];
  const float* b1     = (const float*)d_in[6];
  const float* W2     = (const float*)d_in[7];
  const float* b2     = (const float*)d_in[8];
  const float* gWi    = (const float*)d_in[9];
  const float* gbi    = (const float*)d_in[10];
  const float* gWh    = (const float*)d_in[11];
  const float* gbh    = (const float*)d_in[12];
  const float* Wp1    = (const float*)d_in[13];
  const float* bp1    = (const float*)d_in[14];
  const float* Wp2    = (const float*)d_in[15];
  const float* bp2    = (const float*)d_in[16];
  float* out = (float*)d_out;

  // Workspace carve-up (~38 MB total)
  char* ws = (char*)d_ws;
  const size_t SZ_RW  = (size_t)NN * MSG * sizeof(float);   // 1 MB
  const size_t SZ_HID = (size_t)NN * HID * sizeof(float);   // ~34.6 MB
  const size_t SZ_MSG = (size_t)NN * MSG * sizeof(float);   // 1 MB
  const size_t SZ_UPD = (size_t)NN * MD  * sizeof(float);   // 1 MB
  float* rw   = (float*)(ws);
  float* hid  = (float*)(ws + SZ_RW);
  float* msg  = (float*)(ws + SZ_RW + SZ_HID);
  float* upd  = (float*)(ws + SZ_RW + SZ_HID + SZ_MSG);
  float* pbuf = (float*)(ws + SZ_RW + SZ_HID + SZ_MSG + SZ_UPD);

  // 1) random-walk embedding
  k_rw<<<NN, 64, 0, stream>>>(memory, walks, W_rw, b_rw, rw);
  // 2) big MLP layer 1 (dominant GEMM, f32 WMMA)
  {
    dim3 grid(HID / 64, NN / 64);
    k_gemm1<<<grid, 32, 0, stream>>>(memory, od_mat, rw, W1, b1, hid);
  }
  // 3) MLP layer 2 -> msg
  k_gemm2<<<NN / 16, 32, 0, stream>>>(hid, W2, b2, msg);
  // 4) GRU memory update
  k_gru<<<NN, 64, 0, stream>>>(msg, memory, gWi, gbi, gWh, gbh, upd);
  // 5) prediction hidden layer
  k_pred1<<<NN, 64, 0, stream>>>(upd, Wp1, bp1, pbuf);
  // 6) prediction output GEMM -> od matrix
  {
    dim3 grid(NN / 256, NN / 64);
    k_pred2<<<grid, 128, 0, stream>>>(pbuf, Wp2, bp2, out);
  }
}